// SparseMaskedLinear_v3_11785390260546
// MI455X (gfx1250) — compile-verified
//
#include <hip/hip_runtime.h>
#include <hip/hip_bf16.h>

// Problem constants (from reference)
#define BATCH       64
#define IN_F        20000
#define OUT_F       2000          // also the src-index range (indices < 2000)
#define K_DIM       2000          // effective reduction dim (src < 2000)
#define KSPLIT      5
#define KCHUNK      (K_DIM / KSPLIT)   // 400 columns -> 100 k-steps of 4

typedef __attribute__((ext_vector_type(2))) float v2f;
typedef __attribute__((ext_vector_type(8))) float v8f;
typedef __attribute__((ext_vector_type(4))) float v4f;

// ---------------------------------------------------------------------------
// Kernel 1: zero the dense weight matrix Wd (2000x2000 f32 = 16 MB) in d_ws
// ---------------------------------------------------------------------------
__global__ void k_zero_wd(float* __restrict__ wd, int n4) {
    int i = blockIdx.x * blockDim.x + threadIdx.x;
    if (i < n4) {
        v4f z = {};
        *(v4f*)(wd + (size_t)i * 4) = z;
    }
}

// ---------------------------------------------------------------------------
// Kernel 2: out[b, d] = bias[d]   (output pre-init; GEMM partials atomic-add)
// ---------------------------------------------------------------------------
__global__ void k_init_out(float* __restrict__ out, const float* __restrict__ bias, int n) {
    int i = blockIdx.x * blockDim.x + threadIdx.x;
    if (i < n) out[i] = bias[i % OUT_F];
}

// ---------------------------------------------------------------------------
// Kernel 3: densify edges: Wd[dst, src] += w   (2M L2-resident atomics)
// mask layout: [e][0]=src, [e][1]=dst (int64)
// ---------------------------------------------------------------------------
__global__ void k_scatter_wd(const long long* __restrict__ mask,
                             const float* __restrict__ w,
                             float* __restrict__ wd, int nEdges) {
    int e = blockIdx.x * blockDim.x + threadIdx.x;
    if (e < nEdges) {
        int s = (int)mask[2 * (size_t)e + 0];
        int d = (int)mask[2 * (size_t)e + 1];
        atomicAdd(&wd[(size_t)d * K_DIM + s], w[e]);
    }
}

// ---------------------------------------------------------------------------
// Kernel 4: out += in[:, :2000] x Wd^T via V_WMMA_F32_16X16X4_F32.
// One wave (32 threads) per (Mtile, Ntile, Kchunk). EXEC all-ones, uniform flow.
//
// A (16x4, M=batch rows): lane m=lane%16, hk=lane/16 -> float2 at cols {2hk,2hk+1}
// B (4x16, N=neurons):    B[k][n]=Wd[n][k]          -> float2 from Wd row n
// C/D (v8f): elem v -> M = 8*hk + v, N = lane%16
// ---------------------------------------------------------------------------
__global__ void __launch_bounds__(32)
k_gemm_wmma(const float* __restrict__ in,
            const float* __restrict__ wd,
            float* __restrict__ out) {
    const int bid = blockIdx.x;                 // 4 * 125 * KSPLIT blocks
    const int mt  = bid / (125 * KSPLIT);
    const int rem = bid % (125 * KSPLIT);
    const int nt  = rem / KSPLIT;
    const int kc  = rem % KSPLIT;

    const int lane = threadIdx.x;
    const int nl   = lane & 15;                 // M (for A) or N (for B/C) local
    const int hk   = lane >> 4;                 // which K-pair / which M-half of C

    const float* aPtr = in + (size_t)(mt * 16 + nl) * IN_F + kc * KCHUNK + 2 * hk;
    const float* bPtr = wd + (size_t)(nt * 16 + nl) * K_DIM + kc * KCHUNK + 2 * hk;

    v8f c = {};
#pragma unroll 4
    for (int k = 0; k < KCHUNK; k += 4) {
        v2f a = *(const v2f*)(aPtr + k);
        v2f b = *(const v2f*)(bPtr + k);
        // D = A x B + C  (fp32, 16x16x4)
        c = __builtin_amdgcn_wmma_f32_16x16x4_f32(
                /*neg_a=*/false, a, /*neg_b=*/false, b,
                /*c_mod=*/(short)0, c, /*reuse_a=*/false, /*reuse_b=*/false);
    }

    const int nGlobal = nt * 16 + nl;
    const int mBase   = mt * 16 + hk * 8;
#pragma unroll
    for (int v = 0; v < 8; ++v) {
        atomicAdd(&out[(size_t)(mBase + v) * OUT_F + nGlobal], c[v]);
    }
}

// ---------------------------------------------------------------------------
// Fallback (ws too small): direct per-(edge,batch) atomic scatter. Slow but
// correct (float atomic order is not deterministic).
// ---------------------------------------------------------------------------
__global__ void k_fallback_scatter(const float* __restrict__ in,
                                   const long long* __restrict__ mask,
                                   const float* __restrict__ w,
                                   float* __restrict__ out,
                                   long long total) {
    long long idx = (long long)blockIdx.x * blockDim.x + threadIdx.x;
    long long stride = (long long)gridDim.x * blockDim.x;
    for (; idx < total; idx += stride) {
        long long e = idx >> 6;          // /64
        int b = (int)(idx & 63);
        int s = (int)mask[2 * e + 0];
        int d = (int)mask[2 * e + 1];
        atomicAdd(&out[(size_t)b * OUT_F + d], in[(size_t)b * IN_F + s] * w[e]);
    }
}

extern "C" void kernel_launch(void* const* d_in, const int* in_sizes, int n_in,
                              void* d_out, int out_size, void* d_ws, size_t ws_size,
                              hipStream_t stream) {
    const float*     in   = (const float*)d_in[0];       // [64, 20000] f32
    const long long* mask = (const long long*)d_in[1];   // [E, 2] int64
    const float*     w    = (const float*)d_in[2];       // [E] f32
    const float*     bias = (const float*)d_in[3];       // [2000] f32
    float* out = (float*)d_out;                          // [64, 2000] f32

    const int nEdges = in_sizes[2];
    const size_t WD_BYTES = (size_t)OUT_F * K_DIM * sizeof(float); // 16 MB

    // out = bias broadcast (also zeroes garbage)
    {
        int n = BATCH * OUT_F;
        k_init_out<<<(n + 255) / 256, 256, 0, stream>>>(out, bias, n);
    }

    if (ws_size >= WD_BYTES) {
        float* wd = (float*)d_ws;
        // 1) Wd = 0
        {
            int n4 = (OUT_F * K_DIM) / 4;
            k_zero_wd<<<(n4 + 255) / 256, 256, 0, stream>>>(wd, n4);
        }
        // 2) Wd[dst, src] += w  (2M atomics, L2 resident)
        k_scatter_wd<<<(nEdges + 255) / 256, 256, 0, stream>>>(mask, w, wd, nEdges);
        // 3) out += in[:, :2000] x Wd^T  (fp32 WMMA, K split 5 ways)
        {
            dim3 grid((BATCH / 16) * (OUT_F / 16) * KSPLIT);  // 4*125*5 = 2500 waves
            k_gemm_wmma<<<grid, 32, 0, stream>>>(in, wd, out);
        }
    } else {
        // Fallback: direct atomic scatter over (edge, batch)
        long long total = (long long)nEdges * BATCH;
        int blocks = 16384;
        k_fallback_scatter<<<blocks, 256, 0, stream>>>(in, mask, w, out, total);
    }
}